// FFM_61306363183618
// MI455X (gfx1250) — compile-verified
//
#include <hip/hip_runtime.h>

// ---------------------------------------------------------------------------
// FFM edge interaction + scatter-add for MI455X (gfx1250).
//   out = node_embed + segment_sum(node[src]*node[tgt]*edge_embed, src)
// Memory-bound: ~230 MB HBM traffic (edge_embed stream dominates); node table
// (12.8 MB) is L2-resident so gathers are cheap. We stage edge_embed tiles
// into LDS with double-buffered async DMA (ASYNCcnt) and scatter with
// no-return global_atomic_add_f32.
// ---------------------------------------------------------------------------

#define AS1 __attribute__((address_space(1)))
#define AS3 __attribute__((address_space(3)))

typedef int v4i_t __attribute__((vector_size(16)));   // b128 payload type

constexpr int DIM    = 64;                 // embedding dim
constexpr int BLOCK  = 256;                // 8 waves (wave32)
constexpr int TILE_E = 64;                 // edges per LDS tile (16 KB)
constexpr int TILES_PER_BLOCK = 8;
constexpr int EPB    = TILE_E * TILES_PER_BLOCK;   // 512 edges per block

// ---- CDNA5 async global->LDS copy (b128) with fallbacks -------------------
#if defined(__gfx1250__)
#  if __has_builtin(__builtin_amdgcn_global_load_async_to_lds_b128)
#    define ASYNC_COPY_B128(gp, lp)                                          \
       __builtin_amdgcn_global_load_async_to_lds_b128(                       \
           (AS1 v4i_t*)(gp), (AS3 v4i_t*)(lp), 0, 0)
#  else
#    define ASYNC_COPY_B128(gp, lp)                                          \
       asm volatile("global_load_async_to_lds_b128 %0, %1, off"              \
                    :: "v"((unsigned)(unsigned long long)(lp)),              \
                       "v"((unsigned long long)(gp))                         \
                    : "memory")
#  endif
#  if __has_builtin(__builtin_amdgcn_s_wait_asynccnt)
#    define ASYNC_WAIT(n) __builtin_amdgcn_s_wait_asynccnt(n)
#  else
#    define ASYNC_WAIT(n) asm volatile("s_wait_asynccnt " #n ::: "memory")
#  endif
#else
// Host pass / other archs: synchronous copy so the TU always parses+compiles.
#  define ASYNC_COPY_B128(gp, lp) (*(float4*)(lp) = *(const float4*)(gp))
#  define ASYNC_WAIT(n) ((void)0)
#endif

// ---- out = node_embed (also un-poisons d_out) -----------------------------
__global__ __launch_bounds__(BLOCK) void ffm_init_kernel(
    const float* __restrict__ node, float* __restrict__ out, int n4) {
  int i = blockIdx.x * BLOCK + threadIdx.x;
  if (i < n4) {
    reinterpret_cast<float4*>(out)[i] =
        reinterpret_cast<const float4*>(node)[i];
  }
}

// ---- per-edge interaction + atomic scatter --------------------------------
__global__ __launch_bounds__(BLOCK) void ffm_edge_kernel(
    const float* __restrict__ node,   // [N, 64]
    const float* __restrict__ eemb,   // [E, 64]
    const int*   __restrict__ eidx,   // [2, E] (row0 = src, row1 = tgt)
    float* __restrict__ out,          // [N, 64]
    int E) {
  __shared__ float lds[2][TILE_E * DIM];   // 2 x 16 KB double buffer

  const int tid  = threadIdx.x;
  const int sub  = tid & 15;   // which float4 of the 64-wide row
  const int slot = tid >> 4;   // which edge slot (16 per pass)
  const int blockBase = blockIdx.x * EPB;
  const int* __restrict__ srcIdx = eidx;
  const int* __restrict__ tgtIdx = eidx + E;

  // Stage one 64-edge tile of edge_embed into LDS: 1024 b128 chunks,
  // 4 per thread. Tail chunks clamp the global address (data unused).
  auto issue_tile = [&](int tileBase, int buf) {
#pragma unroll
    for (int c = 0; c < 4; ++c) {
      const int chunk = c * BLOCK + tid;              // 0..1023
      const int edge  = tileBase + (chunk >> 4);      // 16 chunks per edge
      const int ce    = (edge < E) ? edge : 0;        // keep address valid
      const float* g  = eemb + (size_t)ce * DIM + (chunk & 15) * 4;
      float* l        = &lds[buf][chunk * 4];
      ASYNC_COPY_B128(g, l);
    }
  };

  int nTiles = (E - blockBase + TILE_E - 1) / TILE_E;
  if (nTiles > TILES_PER_BLOCK) nTiles = TILES_PER_BLOCK;
  if (nTiles <= 0) return;

  issue_tile(blockBase, 0);

  for (int k = 0; k < nTiles; ++k) {
    const int buf      = k & 1;
    const int tileBase = blockBase + k * TILE_E;
    const bool haveNext = (k + 1) < nTiles;

    if (haveNext) {
      issue_tile(tileBase + TILE_E, buf ^ 1);
      // Warm next tile's index rows into cache (global_prefetch_b8).
      if (tid < TILE_E) {
        const int pe = tileBase + TILE_E + tid;
        if (pe < E) {
          __builtin_prefetch(srcIdx + pe, 0, 0);
          __builtin_prefetch(tgtIdx + pe, 0, 0);
        }
      }
      ASYNC_WAIT(4);   // tile k's 4 loads done (in-order), k+1 in flight
    } else {
      ASYNC_WAIT(0);
    }
    __syncthreads();   // tile k visible to all waves

#pragma unroll
    for (int p = 0; p < 4; ++p) {
      const int eInTile = p * 16 + slot;
      const int e = tileBase + eInTile;
      if (e < E) {
        const int s = srcIdx[e];
        const int t = tgtIdx[e];
        const float4 ns =
            *reinterpret_cast<const float4*>(node + (size_t)s * DIM + sub * 4);
        const float4 nt =
            *reinterpret_cast<const float4*>(node + (size_t)t * DIM + sub * 4);
        const float4 ee =
            *reinterpret_cast<const float4*>(&lds[buf][eInTile * DIM + sub * 4]);
        float4 r;
        r.x = ns.x * nt.x * ee.x;
        r.y = ns.y * nt.y * ee.y;
        r.z = ns.z * nt.z * ee.z;
        r.w = ns.w * nt.w * ee.w;
        float* o = out + (size_t)s * DIM + sub * 4;
        // No-return f32 atomics -> global_atomic_add_f32 (STOREcnt only).
        unsafeAtomicAdd(o + 0, r.x);
        unsafeAtomicAdd(o + 1, r.y);
        unsafeAtomicAdd(o + 2, r.z);
        unsafeAtomicAdd(o + 3, r.w);
      }
    }
    __syncthreads();   // everyone done with tile k before its buffer refills
  }
}

extern "C" void kernel_launch(void* const* d_in, const int* in_sizes, int n_in,
                              void* d_out, int out_size, void* d_ws,
                              size_t ws_size, hipStream_t stream) {
  const float* node = (const float*)d_in[0];   // [N*64] f32
  const float* eemb = (const float*)d_in[1];   // [E*64] f32
  const int*   eidx = (const int*)d_in[2];     // [2*E]  int
  float* out = (float*)d_out;

  const int E  = in_sizes[2] / 2;
  const int n4 = in_sizes[0] / 4;              // N*64/4 float4 copies

  ffm_init_kernel<<<(n4 + BLOCK - 1) / BLOCK, BLOCK, 0, stream>>>(node, out, n4);

  const int blocks = (E + EPB - 1) / EPB;
  ffm_edge_kernel<<<blocks, BLOCK, 0, stream>>>(node, eemb, eidx, out, E);
}